// BagAttention_40063454937108
// MI455X (gfx1250) — compile-verified
//
#include <hip/hip_runtime.h>
#include <hip/hip_bf16.h>
#include <math.h>

#define NSUM      131072
#define HDIM      768
#define NBAGS     4096
#define NUM_CLASS 53

typedef __attribute__((ext_vector_type(2))) float v2f;
typedef __attribute__((ext_vector_type(4))) float v4f;
typedef __attribute__((ext_vector_type(8))) float v8f;

// ---------------------------------------------------------------------------
// Kernel 1: segment boundaries. segment_ids is sorted; offsets[b] = first row
// of bag b, offsets[NBAGS] = NSUM.
// ---------------------------------------------------------------------------
__global__ void seg_bounds_kernel(const int* __restrict__ seg,
                                  int* __restrict__ offsets) {
    int j = blockIdx.x * blockDim.x + threadIdx.x;
    if (j >= NSUM) return;
    int s = seg[j];
    if (j == 0) {
        for (int b = 0; b <= s; ++b) offsets[b] = 0;
    } else {
        int sp = seg[j - 1];
        if (s != sp) {
            for (int b = sp + 1; b <= s; ++b) offsets[b] = j;
        }
    }
    if (j == NSUM - 1) {
        for (int b = s + 1; b <= NBAGS; ++b) offsets[b] = NSUM;
    }
}

// ---------------------------------------------------------------------------
// Kernel 2: fused per-bag attention. One 256-thread (8-wave) workgroup per
// bag (bags are contiguous: segment_ids sorted). Pass 1 streams the bag's
// rep rows from HBM with b128 loads (one row per wave, float4 per lane),
// computing dot(rep[j], W[label[bag]]) with a wave32 xor-shuffle reduction.
// Pass 2: block max -> exp -> block sum. Pass 3 re-reads the rows (L2-hot)
// accumulating p_j * rep[j], each thread owning 3 of the 768 columns.
// ---------------------------------------------------------------------------
__global__ void __launch_bounds__(256)
bag_attn_kernel(const float* __restrict__ rep,
                const float* __restrict__ W,
                const int*   __restrict__ label,
                const int*   __restrict__ offsets,
                float*       __restrict__ scores,
                float*       __restrict__ bag_rep) {
    __shared__ float s_w[HDIM];
    __shared__ float s_red[8];

    const int bag   = blockIdx.x;
    const int tid   = threadIdx.x;
    const int lane  = tid & 31;
    const int wave  = tid >> 5;
    const int start = offsets[bag];
    const int end   = offsets[bag + 1];
    const int lab   = label[bag];

    // stage W[label[bag]] into LDS (3 KB), b128 loads
    {
        const v4f* wsrc = (const v4f*)(W + (size_t)lab * HDIM);
        v4f* wdst = (v4f*)s_w;
        for (int q = tid; q < HDIM / 4; q += 256) wdst[q] = wsrc[q];
    }
    __syncthreads();

    // ---- pass 1: scores + running max (one row per wave, b128 per lane) ----
    float wmax = -INFINITY;
    for (int j = start + wave; j < end; j += 8) {
        const float* r = rep + (size_t)j * HDIM;
        float acc = 0.0f;
        #pragma unroll
        for (int i = 0; i < HDIM / 128; ++i) {           // 6 iterations
            const int h = i * 128 + lane * 4;
            v4f rv = *(const v4f*)(r + h);
            v4f wv = *(const v4f*)(s_w + h);
            acc += rv[0] * wv[0] + rv[1] * wv[1] + rv[2] * wv[2] + rv[3] * wv[3];
        }
        #pragma unroll
        for (int off = 16; off; off >>= 1) acc += __shfl_xor(acc, off, 32);
        if (lane == 0) scores[j] = acc;                  // all lanes hold full sum
        wmax = fmaxf(wmax, acc);
    }
    if (lane == 0) s_red[wave] = wmax;
    __threadfence_block();                               // scores[] visible block-wide
    __syncthreads();
    float bmax = -INFINITY;
    #pragma unroll
    for (int i = 0; i < 8; ++i) bmax = fmaxf(bmax, s_red[i]);
    __syncthreads();

    // ---- pass 2: e = exp(score - max), denom = sum(e) ----
    float psum = 0.0f;
    for (int j = start + tid; j < end; j += 256) {
        float e = __expf(scores[j] - bmax);
        scores[j] = e;
        psum += e;
    }
    #pragma unroll
    for (int off = 16; off; off >>= 1) psum += __shfl_xor(psum, off, 32);
    if (lane == 0) s_red[wave] = psum;
    __threadfence_block();
    __syncthreads();
    float denom = 0.0f;
    #pragma unroll
    for (int i = 0; i < 8; ++i) denom += s_red[i];
    const float inv = 1.0f / denom;

    // ---- pass 3: bag_rep[bag] = sum_j p_j * rep[j] (rows are L2-hot) ----
    const int h0 = tid, h1 = tid + 256, h2 = tid + 512;
    float a0 = 0.0f, a1 = 0.0f, a2 = 0.0f;
    for (int j = start; j < end; ++j) {
        const float p = scores[j] * inv;
        const float* r = rep + (size_t)j * HDIM;
        a0 += p * r[h0];
        a1 += p * r[h1];
        a2 += p * r[h2];
    }
    float* br = bag_rep + (size_t)bag * HDIM;
    br[h0] = a0; br[h1] = a1; br[h2] = a2;
}

// ---------------------------------------------------------------------------
// Kernel 3: bag_logits = bag_rep @ W^T + bias via V_WMMA_F32_16X16X4_F32.
// One wave per 16x16 output tile; 8 waves per block cover 8 M-tiles.
// A (16x4 f32) layout: lane L -> row M = L&15, K pair = 2*(L>>4)+{0,1}.
// B (4x16 f32) layout (transpose-symmetric): lane L -> col N = L&15,
// K pair = 2*(L>>4)+{0,1}; B[k][n] = W[n][k].
// C/D layout: lane L -> N = L&15, VGPR v -> M = v + 8*(L>>4).
//
// Out-of-range classes (n >= 53): D column n depends only on B column n,
// and those lanes are never stored -- so we just CLAMP the W row pointer
// (stay in-bounds) and load unconditionally. No zeroing, no EXEC
// divergence: the inner loop is exactly 2x global_load_b64 + v_wmma.
// ---------------------------------------------------------------------------
__global__ void __launch_bounds__(256)
logits_wmma_kernel(const float* __restrict__ bag_rep,
                   const float* __restrict__ W,
                   const float* __restrict__ bias,
                   float*       __restrict__ out) {
    const int tid   = threadIdx.x;
    const int lane  = tid & 31;
    const int wave  = tid >> 5;
    const int m0    = (blockIdx.x * 8 + wave) * 16;   // M tile base
    const int n0    = blockIdx.y * 16;                // N tile base

    const int mlane = lane & 15;
    const int khalf = lane >> 4;            // 0: K={0,1}, 1: K={2,3}
    const int n     = n0 + mlane;
    const int nclam = (n < NUM_CLASS) ? n : (NUM_CLASS - 1);   // in-bounds row

    const float bval = bias[nclam];
    v8f c;
    #pragma unroll
    for (int v = 0; v < 8; ++v) c[v] = bval;

    const float* arow = bag_rep + (size_t)(m0 + mlane) * HDIM + 2 * khalf;
    const float* brow = W + (size_t)nclam * HDIM + 2 * khalf;

    #pragma unroll 4
    for (int k0 = 0; k0 < HDIM; k0 += 4) {
        v2f a  = *(const v2f*)(arow + k0);
        v2f bb = *(const v2f*)(brow + k0);
        // D = A x B + C   (emits v_wmma_f32_16x16x4_f32)
        c = __builtin_amdgcn_wmma_f32_16x16x4_f32(
                /*neg_a=*/false, a, /*neg_b=*/false, bb,
                /*c_mod=*/(short)0, c, /*reuse_a=*/false, /*reuse_b=*/false);
    }

    if (n < NUM_CLASS) {
        #pragma unroll
        for (int v = 0; v < 8; ++v) {
            const int m = m0 + v + 8 * khalf;
            out[(size_t)m * NUM_CLASS + n] = c[v];
        }
    }
}

// ---------------------------------------------------------------------------
extern "C" void kernel_launch(void* const* d_in, const int* in_sizes, int n_in,
                              void* d_out, int out_size, void* d_ws, size_t ws_size,
                              hipStream_t stream) {
    const float* rep   = (const float*)d_in[0];   // (NSUM, H)
    const float* W     = (const float*)d_in[1];   // (NUM_CLASS, H)
    const float* bias  = (const float*)d_in[2];   // (NUM_CLASS,)
    const int*   label = (const int*)  d_in[3];   // (NBAGS,)
    const int*   seg   = (const int*)  d_in[4];   // (NSUM,) sorted
    float*       out   = (float*)d_out;           // (NBAGS, NUM_CLASS)

    // workspace layout
    char* ws = (char*)d_ws;
    int*   offsets = (int*)ws;                                  // (NBAGS+1)
    float* scores  = (float*)(ws + 32768);                      // NSUM
    float* bag_rep = (float*)(ws + 32768 + (size_t)NSUM * 4);   // NBAGS*H

    // 1) segment boundaries
    seg_bounds_kernel<<<(NSUM + 255) / 256, 256, 0, stream>>>(seg, offsets);

    // 2) fused per-bag attention -> bag_rep
    bag_attn_kernel<<<NBAGS, 256, 0, stream>>>(rep, W, label, offsets,
                                               scores, bag_rep);

    // 3) WMMA classifier: grid = (M tiles / 8 waves, ceil(53/16)=4 N tiles)
    dim3 grid3(NBAGS / (16 * 8), (NUM_CLASS + 15) / 16);
    logits_wmma_kernel<<<grid3, 256, 0, stream>>>(bag_rep, W, bias, out);
}